// TorchMHCCoeffs_71416716198105
// MI455X (gfx1250) — compile-verified
//
#include <hip/hip_runtime.h>
#include <hip/hip_bf16.h>
#include <math.h>

typedef float v2f __attribute__((ext_vector_type(2)));
typedef float v4f __attribute__((ext_vector_type(4)));
typedef float v8f __attribute__((ext_vector_type(8)));

#define HC 4
#define KDIM 8192
#define OUTC 24
#define NROWS 16384
#define NSTEP (KDIM / 4)          // 2048 WMMA K-steps
#define TMAX 8
#define RMS_EPS 1e-6f
#define PRE_EPS 0.01f
#define SINK_EPS 1e-6f
#define POST_MULT 2.0f

// ---------------------------------------------------------------------------
// Prep: pack phi [KDIM][24] into WMMA-B fragment order with pad-to-32 zeros
// baked in:  phiB[step][lane][4] = {B0.vgpr0, B0.vgpr1, B1.vgpr0, B1.vgpr1}
// B layout (16x16x4 f32): VGPR j, lane-half h <-> K = 4*step + j + 2h,
// col = lane%16 (+16 for the second accumulator tile). 1 MB total -> L2-hot.
// ---------------------------------------------------------------------------
__global__ __launch_bounds__(256) void pack_phi_kernel(
    const float* __restrict__ phi, float* __restrict__ phiB)
{
  int t = blockIdx.x * blockDim.x + threadIdx.x;   // t in [0, NSTEP*32)
  if (t >= NSTEP * 32) return;
  int lane = t & 31;
  int step = t >> 5;
  int half = lane >> 4;
  int l16  = lane & 15;
  int kk   = 4 * step + 2 * half;

  const float* p0 = phi + (long)kk * OUTC;
  float b0x = p0[l16];
  float b0y = p0[OUTC + l16];
  float b1x = (l16 < 8) ? p0[16 + l16] : 0.f;          // cols 24..31 -> 0
  float b1y = (l16 < 8) ? p0[OUTC + 16 + l16] : 0.f;

  v4f* dst = (v4f*)phiB + t;
  *dst = (v4f){b0x, b0y, b1x, b1y};
}

// ---------------------------------------------------------------------------
// Main: one wave per 16-row tile. fp32 WMMA accumulate of mix = x @ phi with
// the RMS sum-of-squares folded into the same loop (two independent FMA
// chains to avoid VALU_DEP stalls); sigmoid gates + 4x4 Sinkhorn tail per
// row out of LDS.
// ---------------------------------------------------------------------------
__global__ __launch_bounds__(128) void mhc_coeffs_kernel(
    const float* __restrict__ x, const float* __restrict__ phiB,
    const float* __restrict__ bvec, const float* __restrict__ alpha_pre,
    const float* __restrict__ alpha_post, const float* __restrict__ alpha_res,
    float* __restrict__ out)
{
  __shared__ float smix[4][16][25];  // [wave][row][col] (+1 pad)

  const int tid  = threadIdx.x;
  const int wave = tid >> 5;
  const int lane = tid & 31;
  const int half = lane >> 4;   // 0 -> K%4 in {0,1}; 1 -> K%4 in {2,3}
  const int l16  = lane & 15;

  const int  tile    = blockIdx.x * 4 + wave;
  const long rowBase = (long)tile * 16;

  // ---------------- K loop ----------------
  const float* aptr = x + (rowBase + l16) * (long)KDIM + 2 * half;
  const v4f*   bptr = (const v4f*)phiB + lane;

  v8f c0 = {0.f, 0.f, 0.f, 0.f, 0.f, 0.f, 0.f, 0.f};
  v8f c1 = {0.f, 0.f, 0.f, 0.f, 0.f, 0.f, 0.f, 0.f};
  float sq0 = 0.f, sq1 = 0.f;   // two independent chains -> no VALU_DEP stalls

  for (int s = 0; s < NSTEP; ++s) {
    v2f a  = *(const v2f*)aptr;          // {x[row][k+2h], x[row][k+2h+1]}
    v4f bb = *bptr;                      // packed B fragments, coalesced b128
    aptr += 4;
    bptr += 32;
    sq0 = fmaf(a.x, a.x, sq0);
    sq1 = fmaf(a.y, a.y, sq1);
    v2f b0 = {bb.x, bb.y};
    v2f b1 = {bb.z, bb.w};
    c0 = __builtin_amdgcn_wmma_f32_16x16x4_f32(false, a, false, b0,
                                               (short)0, c0, false, false);
    c1 = __builtin_amdgcn_wmma_f32_16x16x4_f32(false, a, false, b1,
                                               (short)0, c1, false, false);
  }
  float sq = sq0 + sq1;

  // ---------------- per-row 1/rms ----------------
  sq += __shfl_xor(sq, 16, 32);                              // combine K-halves
  float invr = rsqrtf(sq * (1.0f / (float)KDIM) + RMS_EPS);  // lane l: row l%16

  // C/D layout: VGPR g, lanes 0-15 -> M=g, lanes 16-31 -> M=g+8; N=l16
#pragma unroll
  for (int g = 0; g < 8; ++g) {
    int row = g + 8 * half;
    float rv = __shfl(invr, row, 32);
    smix[wave][row][l16] = c0[g] * rv;
    if (l16 < 8) smix[wave][row][16 + l16] = c1[g] * rv;
  }
  __syncthreads();

  // ---------------- per-row tail: gates + Sinkhorn ----------------
  if (lane < 16) {
    const int  r  = lane;
    const long R  = rowBase + r;
    const float ap  = alpha_pre[0];
    const float apo = alpha_post[0];
    const float ar  = alpha_res[0];
    const float* mr = smix[wave][r];

#pragma unroll
    for (int n = 0; n < HC; ++n) {
      float pre  = 1.0f / (1.0f + __expf(-(mr[n] * ap + bvec[n]))) + PRE_EPS;
      out[R * HC + n] = pre;
      float post = (1.0f / (1.0f + __expf(-(mr[HC + n] * apo + bvec[HC + n])))) * POST_MULT;
      out[(long)NROWS * HC + R * HC + n] = post;
    }

    // softmax rows (+eps) -> col norm -> 7x (row norm, col norm)
    float m[4][4];
#pragma unroll
    for (int i = 0; i < 4; ++i) {
      float lg[4], mx = -1e30f;
#pragma unroll
      for (int j = 0; j < 4; ++j) {
        lg[j] = mr[2 * HC + i * 4 + j] * ar + bvec[2 * HC + i * 4 + j];
        mx = fmaxf(mx, lg[j]);
      }
      float s = 0.f;
#pragma unroll
      for (int j = 0; j < 4; ++j) { lg[j] = __expf(lg[j] - mx); s += lg[j]; }
      float inv = 1.0f / s;
#pragma unroll
      for (int j = 0; j < 4; ++j) m[i][j] = lg[j] * inv + SINK_EPS;
    }
#pragma unroll
    for (int j = 0; j < 4; ++j) {
      float s = m[0][j] + m[1][j] + m[2][j] + m[3][j];
      float inv = 1.0f / (s + SINK_EPS);
#pragma unroll
      for (int i = 0; i < 4; ++i) m[i][j] *= inv;
    }
    for (int t = 0; t < TMAX - 1; ++t) {
#pragma unroll
      for (int i = 0; i < 4; ++i) {
        float s = m[i][0] + m[i][1] + m[i][2] + m[i][3];
        float inv = 1.0f / (s + SINK_EPS);
#pragma unroll
        for (int j = 0; j < 4; ++j) m[i][j] *= inv;
      }
#pragma unroll
      for (int j = 0; j < 4; ++j) {
        float s = m[0][j] + m[1][j] + m[2][j] + m[3][j];
        float inv = 1.0f / (s + SINK_EPS);
#pragma unroll
        for (int i = 0; i < 4; ++i) m[i][j] *= inv;
      }
    }
#pragma unroll
    for (int i = 0; i < 4; ++i)
#pragma unroll
      for (int j = 0; j < 4; ++j)
        out[(long)NROWS * 2 * HC + R * (HC * HC) + i * 4 + j] = m[i][j];
  }
}

extern "C" void kernel_launch(void* const* d_in, const int* in_sizes, int n_in,
                              void* d_out, int out_size, void* d_ws, size_t ws_size,
                              hipStream_t stream) {
  const float* x          = (const float*)d_in[0];
  const float* phi        = (const float*)d_in[1];
  const float* bvec       = (const float*)d_in[2];
  const float* alpha_pre  = (const float*)d_in[3];
  const float* alpha_post = (const float*)d_in[4];
  const float* alpha_res  = (const float*)d_in[5];
  float* out  = (float*)d_out;
  float* phiB = (float*)d_ws;                 // needs NSTEP*32*4 floats = 1 MB

  // 1) pack phi into WMMA-B fragment order (zeros baked in) -> L2-resident
  pack_phi_kernel<<<dim3((NSTEP * 32 + 255) / 256), dim3(256), 0, stream>>>(phi, phiB);

  // 2) fused rmsnorm-GEMM-gates-Sinkhorn; 1024 tiles = 256 blocks x 4 waves
  mhc_coeffs_kernel<<<dim3(NROWS / 64), dim3(128), 0, stream>>>(
      x, phiB, bvec, alpha_pre, alpha_post, alpha_res, out);
}